// DepthGlobalPool_42949672961112
// MI455X (gfx1250) — compile-verified
//
#include <hip/hip_runtime.h>
#include <hip/hip_bf16.h>

// Problem constants (match reference setup_inputs)
constexpr int B    = 16;
constexpr int CIN  = 128;
constexpr int COUT = 64;
constexpr int H    = 192;
constexpr int W    = 192;
constexpr int HW   = H * W;          // 36864
constexpr int NPIX = B * HW;         // 589824
constexpr int NBINS = 10;

// Workspace layout (byte offsets into d_ws)
//   [0,8)        : uint minmax[2]  (flipped-float representation)
//   [64,128)     : int counts[16]
//   [128,8320)   : float fsums[16][128]   (rows 10..15 stay zero = WMMA pad)
//   [8448,12544) : float means[16][64]
//   [16384, +NPIX): uint8 bins[NPIX]
// total ~0.6 MB

typedef float v2f __attribute__((ext_vector_type(2)));
typedef float v8f __attribute__((ext_vector_type(8)));

__device__ __forceinline__ unsigned flipf(float f) {
  unsigned u = __float_as_uint(f);
  return (u & 0x80000000u) ? ~u : (u | 0x80000000u);   // monotone float->uint
}
__device__ __forceinline__ float unflipf(unsigned t) {
  return __uint_as_float((t & 0x80000000u) ? (t & 0x7FFFFFFFu) : ~t);
}
__device__ __forceinline__ unsigned umn(unsigned a, unsigned b){ return a < b ? a : b; }
__device__ __forceinline__ unsigned umx(unsigned a, unsigned b){ return a > b ? a : b; }

// ---------------------------------------------------------------- K0: init ws
__global__ void k_init(unsigned* mm, int* counts, float* fsums, float* means) {
  int t = threadIdx.x;
  if (t == 0) { mm[0] = 0xFFFFFFFFu; mm[1] = 0u; }
  if (t < 16) counts[t] = 0;
  for (int i = t; i < 16 * CIN;  i += blockDim.x) fsums[i] = 0.0f;
  for (int i = t; i < 16 * COUT; i += blockDim.x) means[i] = 0.0f;
}

// ------------------------------------------------------ K1: depth min / max
__global__ void k_minmax(const float* __restrict__ depth, unsigned* mm) {
  __shared__ unsigned smin[256];
  __shared__ unsigned smax[256];
  int t = threadIdx.x;
  unsigned lmin = 0xFFFFFFFFu, lmax = 0u;
  for (int i = blockIdx.x * blockDim.x + t; i < NPIX; i += gridDim.x * blockDim.x) {
    unsigned f = flipf(depth[i]);
    lmin = umn(lmin, f);
    lmax = umx(lmax, f);
  }
  smin[t] = lmin; smax[t] = lmax;
  __syncthreads();
  for (int s = 128; s > 0; s >>= 1) {
    if (t < s) { smin[t] = umn(smin[t], smin[t + s]); smax[t] = umx(smax[t], smax[t + s]); }
    __syncthreads();
  }
  if (t == 0) { atomicMin(&mm[0], smin[0]); atomicMax(&mm[1], smax[0]); }
}

// ------------------------------------------- K2: bin ids (u8) + bin counts
__device__ __forceinline__ int bin_of(float d, float dmin, float width) {
  int bi = (int)floorf((d - dmin) / width);
  return bi < 0 ? 0 : (bi > NBINS - 1 ? NBINS - 1 : bi);
}

__global__ void k_bins(const float* __restrict__ depth, const unsigned* __restrict__ mm,
                       unsigned char* __restrict__ bins, int* __restrict__ counts) {
  __shared__ int hist[NBINS];
  int t = threadIdx.x;
  if (t < NBINS) hist[t] = 0;
  __syncthreads();
  float dmin = unflipf(mm[0]);
  float dmax = unflipf(mm[1]);
  float width = (dmax - dmin) / (float)NBINS;
  int g = blockIdx.x * blockDim.x + t;           // group of 4 pixels; grid sized exactly
  float4 d4 = ((const float4*)depth)[g];
  int b0 = bin_of(d4.x, dmin, width);
  int b1 = bin_of(d4.y, dmin, width);
  int b2 = bin_of(d4.z, dmin, width);
  int b3 = bin_of(d4.w, dmin, width);
  unsigned packed = (unsigned)b0 | ((unsigned)b1 << 8) | ((unsigned)b2 << 16) | ((unsigned)b3 << 24);
  ((unsigned*)bins)[g] = packed;
  atomicAdd(&hist[b0], 1); atomicAdd(&hist[b1], 1);
  atomicAdd(&hist[b2], 1); atomicAdd(&hist[b3], 1);
  __syncthreads();
  if (t < NBINS) atomicAdd(&counts[t], hist[t]);
}

// ------------------------------ K3: per-bin feature channel sums (streaming)
// block = (b, cgroup of 4 channels, half plane). Bin bytes staged in LDS once
// and reused across 4 channels. Per-thread accumulators stride-11 padded:
// 11 is odd -> 32 lanes map to 32 distinct LDS banks (mod 64).
constexpr int CG    = 4;
constexpr int SPLIT = 2;
constexpr int CHUNK = HW / SPLIT;                // 18432

__global__ void k_featsum(const float* __restrict__ features,
                          const unsigned char* __restrict__ bins,
                          float* __restrict__ fsums) {
  __shared__ unsigned char sbins[CHUNK];         // 18 KB
  __shared__ float sacc[256 * 11];               // 11.3 KB
  __shared__ float sred[NBINS * 16];
  int t = threadIdx.x;
  int blk  = blockIdx.x;                         // ((b*32 + cg)*2 + half)
  int half = blk & 1;
  int rest = blk >> 1;
  int cg   = rest & 31;
  int b    = rest >> 5;
  int pixbase = b * HW + half * CHUNK;

  const unsigned* bsrc = (const unsigned*)(bins + pixbase);
  unsigned* bdst = (unsigned*)sbins;
  for (int i = t; i < CHUNK / 4; i += 256) bdst[i] = bsrc[i];
  for (int k = 0; k < 11; ++k) sacc[t * 11 + k] = 0.0f;
  __syncthreads();

  for (int ci = 0; ci < CG; ++ci) {
    int c = cg * CG + ci;
    const float* plane = features + (size_t)(b * CIN + c) * HW + half * CHUNK;
    for (int i = t; i < CHUNK; i += 256)
      sacc[t * 11 + (int)sbins[i]] += plane[i];
    __syncthreads();
    if (t < NBINS * 16) {
      int bin = t >> 4, g = t & 15;
      float s = 0.0f;
      for (int j = g; j < 256; j += 16) s += sacc[j * 11 + bin];
      sred[t] = s;
    }
    __syncthreads();
    if (t < NBINS) {
      float s = 0.0f;
      for (int g = 0; g < 16; ++g) s += sred[t * 16 + g];
      atomicAdd(&fsums[t * CIN + c], s);
    }
    for (int k = 0; k < 11; ++k) sacc[t * 11 + k] = 0.0f;
    __syncthreads();
  }
}

// --------------------- K4: means = (fsums·Wᵀ + cnt·bias) / max(cnt,1)  [WMMA]
// M=16 (10 bins zero-padded), N=64 (4 waves x 16-col tiles), K=128 (steps of 4).
// A 16x4 f32 layout (ISA 7.12.2): lane l holds M=l&15; lane-half selects K-half;
// the 2 VGPRs hold 2 consecutive K. B 4x16 mirrors A. C: VGPR j = rows j / j+8.
__global__ void __launch_bounds__(128)
k_means(const float* __restrict__ wt, const float* __restrict__ bias,
        const float* __restrict__ fsums, const int* __restrict__ counts,
        float* __restrict__ means) {
  int t = threadIdx.x;
  int wave  = t >> 5;
  int lane  = t & 31;
  int cout0 = wave * 16;
  int mrow  = lane & 15;            // A row / B column within tile
  int khalf = (lane >> 4) * 2;      // K-half select
  v8f acc = {};
  for (int k0 = 0; k0 < CIN; k0 += 4) {
    v2f a, bm;
    a.x  = fsums[mrow * CIN + k0 + khalf];
    a.y  = fsums[mrow * CIN + k0 + khalf + 1];
    bm.x = wt[(cout0 + mrow) * CIN + k0 + khalf];      // B[k][n] = weight[n][k]
    bm.y = wt[(cout0 + mrow) * CIN + k0 + khalf + 1];
    acc = __builtin_amdgcn_wmma_f32_16x16x4_f32(
        false, a, false, bm, (short)0, acc, false, false);
  }
  int nabs = cout0 + (lane & 15);
  float bv = bias[nabs];
  #pragma unroll
  for (int j = 0; j < 8; ++j) {
    int m = j + ((lane >> 4) << 3);
    if (m < NBINS) {
      float cnt = (float)counts[m];
      means[m * COUT + nabs] = (acc[j] + cnt * bv) / fmaxf(cnt, 1.0f);
    }
  }
}

// ---------------------------------------- K5: scatter means -> output (f4)
// sm transposed to [c*16 + m]: lanes share c, differ in m<10 -> distinct banks.
__global__ void k_scatter(const unsigned char* __restrict__ bins,
                          const float* __restrict__ means,
                          float* __restrict__ out) {
  __shared__ float sm[COUT * 16];
  int t = threadIdx.x;
  for (int i = t; i < NBINS * COUT; i += 256) {
    int m = i / COUT, c = i % COUT;
    sm[(c << 4) + m] = means[i];
  }
  __syncthreads();
  int g  = blockIdx.x * blockDim.x + t;          // float4 group; grid sized exactly
  int gi = g;
  int w4 = gi % (W / 4); gi /= (W / 4);
  int h  = gi % H;       gi /= H;
  int c  = gi % COUT;
  int b  = gi / COUT;
  int pix = (b * H + h) * W + w4 * 4;
  uchar4 bn = ((const uchar4*)bins)[pix >> 2];
  int cb = c << 4;
  float4 o;
  o.x = sm[cb + bn.x];
  o.y = sm[cb + bn.y];
  o.z = sm[cb + bn.z];
  o.w = sm[cb + bn.w];
  ((float4*)out)[g] = o;
}

// -------------------------------------------------------------------- launch
extern "C" void kernel_launch(void* const* d_in, const int* in_sizes, int n_in,
                              void* d_out, int out_size, void* d_ws, size_t ws_size,
                              hipStream_t stream) {
  const float* features = (const float*)d_in[0];
  const float* depth    = (const float*)d_in[1];
  const float* weight   = (const float*)d_in[2];
  const float* bias     = (const float*)d_in[3];
  float* out = (float*)d_out;

  char* ws = (char*)d_ws;
  unsigned* mm   = (unsigned*)(ws);
  int*      cnts = (int*)(ws + 64);
  float*    fs   = (float*)(ws + 128);
  float*    mns  = (float*)(ws + 8448);
  unsigned char* bins = (unsigned char*)(ws + 16384);

  k_init   <<<1, 256, 0, stream>>>(mm, cnts, fs, mns);
  k_minmax <<<1024, 256, 0, stream>>>(depth, mm);
  k_bins   <<<NPIX / 4 / 256, 256, 0, stream>>>(depth, mm, bins, cnts);   // 576 blocks
  k_featsum<<<B * (CIN / CG) * SPLIT, 256, 0, stream>>>(features, bins, fs); // 1024 blocks
  k_means  <<<1, 128, 0, stream>>>(weight, bias, fs, cnts, mns);
  k_scatter<<<(B * COUT * HW / 4) / 256, 256, 0, stream>>>(bins, mns, out); // 36864 blocks
}